// DetectionLayer_86517821216529
// MI455X (gfx1250) — compile-verified
//
#include <hip/hip_runtime.h>
#include <stdint.h>
#include <math.h>

// Problem constants (match reference)
constexpr int  kB   = 16;       // batch
constexpr int  kN   = 4096;     // rois per image
constexpr int  kC   = 81;       // classes
constexpr int  kK   = 100;      // max instances
constexpr float kNEG  = -1.0e9f;
constexpr float kMINC = 0.7f;
constexpr float kNMST = 0.3f;

constexpr int ROIS_PER_BLOCK = 128;
constexpr int REFINE_BLOCK   = 256;
constexpr int NMS_BLOCK      = 1024;   // 32 wave32 waves; 4 elements/thread

// ---------------------------------------------------------------------------
// CDNA5 async global -> LDS copies (ASYNCcnt path). Guarded so host pass and
// toolchains without the builtins still compile (fall back to plain ld/st).
// b32 signature probe-confirmed: (global int*, local int*, Ii off, Ii cpol).
// ---------------------------------------------------------------------------
#if defined(__HIP_DEVICE_COMPILE__) && defined(__gfx1250__)
#if __has_builtin(__builtin_amdgcn_global_load_async_to_lds_b32)
#define USE_ASYNC_LDS 1
#endif
#if __has_builtin(__builtin_amdgcn_global_load_async_to_lds_b128)
#define USE_ASYNC_LDS_B128 1
#endif
#endif

#ifdef USE_ASYNC_LDS
typedef __attribute__((address_space(1))) int* gint_ptr;
typedef __attribute__((address_space(3))) int* lint_ptr;
#endif
#ifdef USE_ASYNC_LDS_B128
typedef int v4i_t __attribute__((ext_vector_type(4)));
typedef __attribute__((address_space(1))) v4i_t* gv4i_ptr;
typedef __attribute__((address_space(3))) v4i_t* lv4i_ptr;
#endif

// Copy 16 bytes (4 floats) global -> LDS. Both pointers must be 16B aligned.
__device__ __forceinline__ void async_copy_b128(const float* g, float* l) {
#if defined(USE_ASYNC_LDS_B128)
  __builtin_amdgcn_global_load_async_to_lds_b128(
      (gv4i_ptr)(g), (lv4i_ptr)(l), /*offset=*/0, /*cpol=*/0);
#elif defined(USE_ASYNC_LDS)
  __builtin_amdgcn_global_load_async_to_lds_b32((gint_ptr)(g + 0), (lint_ptr)(l + 0), 0, 0);
  __builtin_amdgcn_global_load_async_to_lds_b32((gint_ptr)(g + 1), (lint_ptr)(l + 1), 0, 0);
  __builtin_amdgcn_global_load_async_to_lds_b32((gint_ptr)(g + 2), (lint_ptr)(l + 2), 0, 0);
  __builtin_amdgcn_global_load_async_to_lds_b32((gint_ptr)(g + 3), (lint_ptr)(l + 3), 0, 0);
#else
  *(float4*)l = *(const float4*)g;
#endif
}

__device__ __forceinline__ void wait_async_lds() {
#if defined(USE_ASYNC_LDS) || defined(USE_ASYNC_LDS_B128)
#if __has_builtin(__builtin_amdgcn_s_wait_asynccnt)
  __builtin_amdgcn_s_wait_asynccnt(0);
#else
  asm volatile("s_wait_asynccnt 0" ::: "memory");
#endif
#endif
}

// ---------------------------------------------------------------------------
// Workspace layout (floats):
//   [0, 4*B*N)            refined boxes (float4 per roi: y1,x1,y2,x2)
//   [4BN, 5BN)            class id (as float)
//   [5BN, 6BN)            class score
//   [6BN, 7BN)            masked score (NEG if invalid)
// Total = 7*16*4096*4B = 1.75 MB
// ---------------------------------------------------------------------------

__global__ __launch_bounds__(REFINE_BLOCK)
void refine_kernel(const float* __restrict__ rois,
                   const float* __restrict__ probs,
                   const float* __restrict__ deltas,
                   float* __restrict__ ws) {
  __shared__ __align__(16) float sprob[ROIS_PER_BLOCK * kC];   // 41,472 B

  const int tid = threadIdx.x;
  const long long roi0 = (long long)blockIdx.x * ROIS_PER_BLOCK;

  // Stream this block's prob slab into LDS via the async DMA pipe, 16B per
  // transfer. Slab byte offset = 41472*blockIdx -> 16B aligned; slab size
  // 10368 floats = 2592 float4 chunks.
  constexpr int kChunks = (ROIS_PER_BLOCK * kC) / 4;
  const float* gp = probs + roi0 * kC;
  for (int c = tid; c < kChunks; c += REFINE_BLOCK)
    async_copy_b128(gp + 4 * c, sprob + 4 * c);
  wait_async_lds();
  __syncthreads();

  if (tid < ROIS_PER_BLOCK) {
    const long long roi = roi0 + tid;
    // argmax, first occurrence wins (strict > keeps earliest index).
    // LDS stride 81 == 17 (mod 64 banks), 17 coprime 64 -> conflict-free.
    const float* row = sprob + tid * kC;
    float best = row[0];
    int   cls  = 0;
    for (int j = 1; j < kC; ++j) {
      float v = row[j];
      if (v > best) { best = v; cls = j; }
    }

    // Gather only the winning class's delta (16B aligned float4).
    const float4 d  = ((const float4*)deltas)[roi * kC + cls];
    const float4 rb = ((const float4*)rois)[roi];   // y1,x1,y2,x2

    float h  = rb.z - rb.x;
    float w  = rb.w - rb.y;
    float cy = rb.x + 0.5f * h + (d.x * 0.1f) * h;
    float cx = rb.y + 0.5f * w + (d.y * 0.1f) * w;
    h *= expf(d.z * 0.2f);
    w *= expf(d.w * 0.2f);
    float y1 = fminf(fmaxf(cy - 0.5f * h, 0.0f), 1.0f);
    float x1 = fminf(fmaxf(cx - 0.5f * w, 0.0f), 1.0f);
    float y2 = fminf(fmaxf(cy + 0.5f * h, 0.0f), 1.0f);
    float x2 = fminf(fmaxf(cx + 0.5f * w, 0.0f), 1.0f);

    const long long BN = (long long)kB * kN;
    ((float4*)ws)[roi] = make_float4(y1, x1, y2, x2);
    float* wcls = ws + 4 * BN;
    float* wscr = wcls + BN;
    float* wsc0 = wscr + BN;
    wcls[roi] = (float)cls;
    wscr[roi] = best;
    wsc0[roi] = (cls > 0 && best >= kMINC) ? best : kNEG;
  }
}

// Combine rule matching jnp.argmax first-occurrence semantics.
__device__ __forceinline__ void arg_combine(float& bv, int& bi, float ov, int oi) {
  if (ov > bv || (ov == bv && oi < bi)) { bv = ov; bi = oi; }
}

__global__ __launch_bounds__(NMS_BLOCK)
void nms_kernel(const float* __restrict__ ws, float* __restrict__ out) {
  const int tid  = threadIdx.x;
  const int b    = blockIdx.x;
  const int lane = tid & 31;
  const int wid  = tid >> 5;

  __shared__ float rv[32];
  __shared__ int   ri[32];
  __shared__ float bc[4];
  __shared__ int   sel_idx[kK];
  __shared__ float sel_scr[kK];

  const long long BN   = (long long)kB * kN;
  const float* wbox = ws;
  const float* wcls = ws + 4 * BN;
  const float* wscr = wcls + BN;
  const float* wsc0 = wscr + BN;
  const long long base = (long long)b * kN;

  // 4096 candidates held entirely in registers: 4 per thread, contiguous.
  float sc[4], oy1[4], ox1[4], oy2[4], ox2[4];
  #pragma unroll
  for (int e = 0; e < 4; ++e) {
    const int j = tid * 4 + e;
    const float4 bx = ((const float4*)wbox)[base + j];
    const float off = 2.0f * wcls[base + j];     // per-class NMS offset trick
    oy1[e] = bx.x + off; ox1[e] = bx.y + off;
    oy2[e] = bx.z + off; ox2[e] = bx.w + off;
    sc[e]  = wsc0[base + j];
  }

  for (int k = 0; k < kK; ++k) {
    // ---- block argmax: registers -> wave32 shuffle -> 32 LDS partials ----
    float bv = -3.402823466e38f;
    int   bi = 0x7fffffff;
    #pragma unroll
    for (int e = 0; e < 4; ++e)
      if (sc[e] > bv) { bv = sc[e]; bi = tid * 4 + e; }   // in-order: first max

    #pragma unroll
    for (int off = 16; off > 0; off >>= 1) {
      float ov = __shfl_down(bv, (unsigned)off, 32);
      int   oi = __shfl_down(bi, (unsigned)off, 32);
      arg_combine(bv, bi, ov, oi);
    }
    if (lane == 0) { rv[wid] = bv; ri[wid] = bi; }
    __syncthreads();
    if (wid == 0) {
      bv = rv[lane];
      bi = ri[lane];
      #pragma unroll
      for (int off = 16; off > 0; off >>= 1) {
        float ov = __shfl_down(bv, (unsigned)off, 32);
        int   oi = __shfl_down(bi, (unsigned)off, 32);
        arg_combine(bv, bi, ov, oi);
      }
      if (lane == 0) { rv[0] = bv; ri[0] = bi; sel_idx[k] = bi; sel_scr[k] = bv; }
    }
    __syncthreads();

    // ---- broadcast winner's offset box; owner marks it suppressed ----
    const int i = ri[0];
    if (tid == (i >> 2)) {
      const int e = i & 3;
      bc[0] = oy1[e]; bc[1] = ox1[e]; bc[2] = oy2[e]; bc[3] = ox2[e];
      sc[e] = kNEG;
    }
    __syncthreads();

    // ---- register-resident IoU suppression ----
    const float ay1 = bc[0], ax1 = bc[1], ay2 = bc[2], ax2 = bc[3];
    const float area_a = (ay2 - ay1) * (ax2 - ax1);
    #pragma unroll
    for (int e = 0; e < 4; ++e) {
      float iy1 = fmaxf(ay1, oy1[e]);
      float ix1 = fmaxf(ax1, ox1[e]);
      float iy2 = fminf(ay2, oy2[e]);
      float ix2 = fminf(ax2, ox2[e]);
      float inter  = fmaxf(iy2 - iy1, 0.0f) * fmaxf(ix2 - ix1, 0.0f);
      float area_b = (oy2[e] - oy1[e]) * (ox2[e] - ox1[e]);
      float uni    = fmaxf(area_a + area_b - inter, 1e-10f);
      if (inter / uni > kNMST) sc[e] = kNEG;
    }
    // next iteration's LDS writes are ordered by its own __syncthreads
  }

  __syncthreads();
  if (tid < kK) {
    const int   i  = sel_idx[tid];
    const float s  = sel_scr[tid];
    const float ok = (s > 0.5f * kNEG) ? 1.0f : 0.0f;   // s > -5e8
    const float4 bx = ((const float4*)wbox)[base + i];
    const float cls = wcls[base + i];
    const float scr = wscr[base + i];
    float* o = out + ((long long)b * kK + tid) * 6;
    o[0] = bx.x * ok; o[1] = bx.y * ok; o[2] = bx.z * ok; o[3] = bx.w * ok;
    o[4] = cls * ok;  o[5] = scr * ok;
  }
}

extern "C" void kernel_launch(void* const* d_in, const int* in_sizes, int n_in,
                              void* d_out, int out_size, void* d_ws, size_t ws_size,
                              hipStream_t stream) {
  const float* rois   = (const float*)d_in[0];
  const float* probs  = (const float*)d_in[1];
  const float* deltas = (const float*)d_in[2];
  float* ws  = (float*)d_ws;    // needs 7*B*N*4 = 1.75 MB
  float* out = (float*)d_out;   // [16,100,6] f32

  refine_kernel<<<(kB * kN) / ROIS_PER_BLOCK, REFINE_BLOCK, 0, stream>>>(
      rois, probs, deltas, ws);
  nms_kernel<<<kB, NMS_BLOCK, 0, stream>>>(ws, out);
}